// GraphDeConvolution_18528488915635
// MI455X (gfx1250) — compile-verified
//
#include <hip/hip_runtime.h>

typedef float v2f __attribute__((ext_vector_type(2)));
typedef float v8f __attribute__((ext_vector_type(8)));

#define N_NODES 50000
#define D_IN    512
#define F_OUT   256
#define BM      64       // rows per block
#define KC      64       // K-chunk staged in LDS
#define SX_STRIDE 68     // mult of 4: keeps b128 staging stores 16B aligned; bank shift 4/row
#define SW_PITCH  272    // pair-row pitch: lane halves land in disjoint bank halves

#if __has_builtin(__builtin_amdgcn_global_load_async_to_lds_b128) && \
    __has_builtin(__builtin_amdgcn_s_wait_asynccnt)
#define HAS_ASYNC_LDS 1
typedef int v4i __attribute__((vector_size(16)));   // matches builtin's param type
typedef __attribute__((address_space(1))) v4i* gv4i_ptr;
typedef __attribute__((address_space(3))) v4i* lv4i_ptr;
#else
#define HAS_ASYNC_LDS 0
#endif

// ---------------------------------------------------------------------------
// Kernel 1: support[branch] = X[branch] @ W  via V_WMMA_F32_16X16X4_F32
//   256 threads = 8 waves; wave w owns row-tile (w>>1) and column half (w&1).
//   sW holds W k-pair-interleaved: element (k,n) at ((k>>1)*SW_PITCH + n)*2 + (k&1)
//   so each lane's B fragment {W[k][n], W[k+1][n]} is ONE aligned ds_load_b64.
// ---------------------------------------------------------------------------
__global__ __launch_bounds__(256) void gemm_wmma_f32(
    const float* __restrict__ Xori,
    const float* __restrict__ Xaug,
    const float* __restrict__ W,
    float* __restrict__ supp)
{
    __shared__ float sX[BM * SX_STRIDE];          // 17.4 KB
    __shared__ float sW[(KC / 2) * SW_PITCH * 2]; // 69.6 KB

    const int tid  = threadIdx.x;
    const int wave = tid >> 5;             // 0..7
    const int lane = tid & 31;
    const int rt   = wave >> 1;            // row tile 0..3 (16 rows each)
    const int ch   = wave & 1;             // column half: tiles [ch*8, ch*8+8)
    const int baseRow = blockIdx.x * BM;
    const int branch  = blockIdx.y;
    const float* __restrict__ X = branch ? Xaug : Xori;
    float* __restrict__ out = supp + (size_t)branch * N_NODES * F_OUT;

    // ISA 7.12.2 layouts for 16x16x4 f32:
    //   A 16x4 : lanes 0-15 -> K={0,1}; lanes 16-31 -> K={2,3}  (2 VGPRs)
    //   B 4x16 : VGPR v: lanes 0-15 row K=v, lanes 16-31 row K=v+2
    //   C 16x16: VGPR v: lanes 0-15 row M=v, lanes 16-31 row M=v+8
    const int halfSel = lane >> 4;         // 0 | 1
    const int m16     = lane & 15;
    const int kOff    = halfSel * 2;

    v8f acc[8] = {};                       // 8 column tiles x 8 VGPRs = 64 VGPRs

    const float* aBase = sX + (rt * 16 + m16) * SX_STRIDE + kOff;
    // B base: pair index = ks*2 + halfSel covers rows {4ks+kOff, 4ks+kOff+1}
    const float* bCol  = sW + (ch * 128 + m16) * 2;

    for (int kc = 0; kc < D_IN; kc += KC) {
        // ---- stage X tile [BM x KC]: 4096 floats ----
        {
            const int r0   = tid >> 4;         // 0..15
            const int colV = (tid & 15) * 4;   // float4 column
            for (int i = 0; i < 4; ++i) {
                const int r  = r0 + i * 16;
                int gr = baseRow + r;
                gr = gr < N_NODES ? gr : N_NODES - 1;   // clamp; bad rows masked at store
                const float* src = X + (size_t)gr * D_IN + kc + colV;
                float* dst = sX + r * SX_STRIDE + colV;
#if HAS_ASYNC_LDS
                // GLOBAL_LOAD_ASYNC_TO_LDS_B128: memory -> LDS, no VGPR round-trip
                __builtin_amdgcn_global_load_async_to_lds_b128(
                    (gv4i_ptr)(v4i*)src, (lv4i_ptr)dst, 0, 0);
#else
                const float4 v = *(const float4*)src;
                dst[0] = v.x; dst[1] = v.y; dst[2] = v.z; dst[3] = v.w;
#endif
            }
        }
        // ---- stage W tile [KC x F_OUT], k-pair interleaved (v2f ds stores) ----
        {
            // 32 pairs x 64 float4-cols = 2048 tasks, 8 per thread
            for (int i = 0; i < 8; ++i) {
                const int t    = tid + i * 256;
                const int pair = t >> 6;            // 0..31
                const int c4   = (t & 63) * 4;
                const float4 r0 = *(const float4*)(W + (size_t)(kc + 2 * pair) * F_OUT + c4);
                const float4 r1 = *(const float4*)(W + (size_t)(kc + 2 * pair + 1) * F_OUT + c4);
                float* dst = sW + (pair * SW_PITCH + c4) * 2;
                *(v2f*)(dst + 0) = v2f{r0.x, r1.x};
                *(v2f*)(dst + 2) = v2f{r0.y, r1.y};
                *(v2f*)(dst + 4) = v2f{r0.z, r1.z};
                *(v2f*)(dst + 6) = v2f{r0.w, r1.w};
            }
        }
#if HAS_ASYNC_LDS
        __builtin_amdgcn_s_wait_asynccnt(0);
#endif
        __syncthreads();

        // ---- 16 k-steps; one A frag live, 8 independent acc chains per step ----
        #pragma unroll
        for (int ks = 0; ks < 16; ++ks) {
            const v2f a = *(const v2f*)(aBase + ks * 4);               // ds_load_b64
            const float* bBase = bCol + (ks * 2 + halfSel) * (SW_PITCH * 2);
            #pragma unroll
            for (int ct = 0; ct < 8; ++ct) {
                const v2f b = *(const v2f*)(bBase + ct * 32);          // ds_load_b64
                acc[ct] = __builtin_amdgcn_wmma_f32_16x16x4_f32(
                    false, a, false, b, (short)0, acc[ct], false, false);
            }
        }
        __syncthreads();
    }

    // ---- store C tiles (guard partial last block: 50000 = 781*64 + 16) ----
    const int rowHalf = halfSel * 8;
    #pragma unroll
    for (int ct = 0; ct < 8; ++ct) {
        #pragma unroll
        for (int v = 0; v < 8; ++v) {
            const int gr = baseRow + rt * 16 + v + rowHalf;
            if (gr < N_NODES)
                out[(size_t)gr * F_OUT + (ch * 8 + ct) * 16 + m16] = acc[ct][v];
        }
    }
}

// ---------------------------------------------------------------------------
// Kernel 2: COO scatter-add for both branches (f32 atomics land in L2;
//           both support arrays fit in the 192 MB L2)
// ---------------------------------------------------------------------------
__global__ __launch_bounds__(256) void edge_scatter(
    const float* __restrict__ supp,
    const int*   __restrict__ eRow,
    const int*   __restrict__ eCol,
    const float* __restrict__ eVal,
    float* __restrict__ out, int E)
{
    const int g = threadIdx.x >> 6;        // 4 edges per block
    const int l = threadIdx.x & 63;        // 64 threads x float4 = 256 floats
    const int e = blockIdx.x * 4 + g;
    if (e >= E) return;

    const int   r = eRow[e];
    const int   c = eCol[e];
    const float v = eVal[e];
    const size_t stride = (size_t)N_NODES * F_OUT;
    const size_t fo = (size_t)l * 4;

    const float4 s0 = *(const float4*)(supp + (size_t)c * F_OUT + fo);
    const float4 s1 = *(const float4*)(supp + stride + (size_t)c * F_OUT + fo);
    float* o0 = out + (size_t)r * F_OUT + fo;
    float* o1 = o0 + stride;

    atomicAdd(o0 + 0, v * s0.x); atomicAdd(o0 + 1, v * s0.y);
    atomicAdd(o0 + 2, v * s0.z); atomicAdd(o0 + 3, v * s0.w);
    atomicAdd(o1 + 0, v * s1.x); atomicAdd(o1 + 1, v * s1.y);
    atomicAdd(o1 + 2, v * s1.z); atomicAdd(o1 + 3, v * s1.w);
}

// ---------------------------------------------------------------------------
// Kernel 3: out = relu(out + bias), in place, float4
// ---------------------------------------------------------------------------
__global__ __launch_bounds__(256) void bias_relu(float* __restrict__ out,
                                                 const float* __restrict__ bias,
                                                 size_t nVec4)
{
    const size_t i = (size_t)blockIdx.x * blockDim.x + threadIdx.x;
    if (i >= nVec4) return;
    const float4 b = *(const float4*)(bias + (i & 63) * 4);  // F_OUT/4 = 64
    float4 x = *(float4*)(out + i * 4);
    x.x = fmaxf(x.x + b.x, 0.f);
    x.y = fmaxf(x.y + b.y, 0.f);
    x.z = fmaxf(x.z + b.z, 0.f);
    x.w = fmaxf(x.w + b.w, 0.f);
    *(float4*)(out + i * 4) = x;
}

__global__ __launch_bounds__(256) void zero_f4(float* __restrict__ p, size_t nVec4)
{
    const size_t i = (size_t)blockIdx.x * blockDim.x + threadIdx.x;
    if (i < nVec4) *(float4*)(p + i * 4) = make_float4(0.f, 0.f, 0.f, 0.f);
}

// ---------------------------------------------------------------------------
extern "C" void kernel_launch(void* const* d_in, const int* in_sizes, int n_in,
                              void* d_out, int out_size, void* d_ws, size_t ws_size,
                              hipStream_t stream)
{
    const float* Xori = (const float*)d_in[0];
    const float* Xaug = (const float*)d_in[1];
    const int*   eRow = (const int*)  d_in[2];
    const int*   eCol = (const int*)  d_in[3];
    const float* eVal = (const float*)d_in[4];
    const float* W    = (const float*)d_in[5];
    const float* bias = (const float*)d_in[6];
    float* out  = (float*)d_out;
    float* supp = (float*)d_ws;            // 2 x 50000 x 256 f32 = 102.4 MB

    const int E = in_sizes[2];
    const size_t nVec4 = (2ull * N_NODES * F_OUT) / 4;

    zero_f4<<<dim3((unsigned)((nVec4 + 255) / 256)), 256, 0, stream>>>(out, nVec4);

    gemm_wmma_f32<<<dim3((N_NODES + BM - 1) / BM, 2), 256, 0, stream>>>(
        Xori, Xaug, W, supp);

    edge_scatter<<<dim3((unsigned)((E + 3) / 4)), 256, 0, stream>>>(
        supp, eRow, eCol, eVal, out, E);

    bias_relu<<<dim3((unsigned)((nVec4 + 255) / 256)), 256, 0, stream>>>(
        out, bias, nVec4);
}